// CompGCNLayer_8254927142972
// MI455X (gfx1250) — compile-verified
//
#include <hip/hip_runtime.h>

typedef float v2f __attribute__((ext_vector_type(2)));
typedef float v8f __attribute__((ext_vector_type(8)));

#define DD 128
#define LDS_PITCH 132   // 128 + 4 pad: A-frag column reads hit distinct banks

// ---------------------------------------------------------------------------
// Kernel 1: hn = h * norm (row-broadcast), pre = 0
// ---------------------------------------------------------------------------
__global__ void compgcn_prep_kernel(const float* __restrict__ h,
                                    const float* __restrict__ norm,
                                    float* __restrict__ hn,
                                    float* __restrict__ pre,
                                    int total4) {
    int i = blockIdx.x * blockDim.x + threadIdx.x;
    if (i >= total4) return;
    int row = i >> 5;                       // 32 float4 chunks per 128-float row
    float nv = norm[row];
    float4 v = ((const float4*)h)[i];
    v.x *= nv; v.y *= nv; v.z *= nv; v.w *= nv;
    ((float4*)hn)[i] = v;
    ((float4*)pre)[i] = make_float4(0.f, 0.f, 0.f, 0.f);
}

// ---------------------------------------------------------------------------
// Kernel 2: pre[dst] += hn[src] - r[rel]   (one wave per edge, 4 floats/lane)
// ---------------------------------------------------------------------------
__global__ void compgcn_edge_kernel(const float* __restrict__ hn,
                                    const float* __restrict__ r,
                                    const int* __restrict__ src,
                                    const int* __restrict__ dst,
                                    const int* __restrict__ rel,
                                    float* __restrict__ pre,
                                    int E) {
    int idx = blockIdx.x * blockDim.x + threadIdx.x;
    int e = idx >> 5;
    if (e >= E) return;
    int c = (idx & 31) << 2;                // this lane's 4-float chunk
    int s  = src[e];
    int d  = dst[e];
    int rr = rel[e];
    float4 hv = *(const float4*)(hn + (size_t)s  * DD + c);
    float4 rv = *(const float4*)(r  + (size_t)rr * DD + c);
    float* p = pre + (size_t)d * DD + c;
    __hip_atomic_fetch_add(p + 0, hv.x - rv.x, __ATOMIC_RELAXED, __HIP_MEMORY_SCOPE_AGENT);
    __hip_atomic_fetch_add(p + 1, hv.y - rv.y, __ATOMIC_RELAXED, __HIP_MEMORY_SCOPE_AGENT);
    __hip_atomic_fetch_add(p + 2, hv.z - rv.z, __ATOMIC_RELAXED, __HIP_MEMORY_SCOPE_AGENT);
    __hip_atomic_fetch_add(p + 3, hv.w - rv.w, __ATOMIC_RELAXED, __HIP_MEMORY_SCOPE_AGENT);
}

// ---------------------------------------------------------------------------
// Kernel 3: out = relu(hn @ W + (pre @ W_msg) * norm + b)
// Block: 256 threads = 8 waves; block owns 16 node rows; wave w owns cols
// [16w, 16w+16). K-loop step 4 -> v_wmma_f32_16x16x4_f32 (two per step).
// ---------------------------------------------------------------------------
__global__ __launch_bounds__(256) void compgcn_gemm_kernel(
        const float* __restrict__ hn,
        const float* __restrict__ pre,
        const float* __restrict__ norm,
        const float* __restrict__ W,
        const float* __restrict__ Wmsg,
        const float* __restrict__ b,
        float* __restrict__ out,
        int N) {
    __shared__ float lds_hn [16 * LDS_PITCH];
    __shared__ float lds_pre[16 * LDS_PITCH];

    const int nodeBase = blockIdx.x * 16;
    const int t = threadIdx.x;
    const bool fullTile = (nodeBase + 16 <= N);

    // Cooperative stage of the 16x128 hn/pre tiles into LDS (float4, padded rows)
    for (int f = t; f < 512; f += 256) {           // 512 float4 per tile
        int row = f >> 5;
        int c4  = (f & 31) << 2;
        int g   = nodeBase + row;
        float4 hv = make_float4(0.f, 0.f, 0.f, 0.f);
        float4 pv = make_float4(0.f, 0.f, 0.f, 0.f);
        if (g < N) {
            hv = *(const float4*)(hn  + (size_t)g * DD + c4);
            pv = *(const float4*)(pre + (size_t)g * DD + c4);
        }
        *(float4*)&lds_hn [row * LDS_PITCH + c4] = hv;
        *(float4*)&lds_pre[row * LDS_PITCH + c4] = pv;
    }
    __syncthreads();

    const int wave = t >> 5;
    const int lane = t & 31;
    const int colBase = wave << 4;
    const int m  = lane & 15;       // A-frag row / B-frag column index
    const int kh = lane >> 4;       // lane half selects K pair {0,1} vs {2,3}
    const int col = colBase + m;

    v8f accW = {0.f, 0.f, 0.f, 0.f, 0.f, 0.f, 0.f, 0.f};
    v8f accM = {0.f, 0.f, 0.f, 0.f, 0.f, 0.f, 0.f, 0.f};

    #pragma unroll 4
    for (int k0 = 0; k0 < DD; k0 += 4) {
        const int k = k0 + 2 * kh;
        // A fragment (16x4): lanes 0-15 -> K={k0,k0+1}, lanes 16-31 -> K={k0+2,k0+3}
        v2f a, bf, a2, b2;
        a[0]  = lds_hn[m * LDS_PITCH + k];
        a[1]  = lds_hn[m * LDS_PITCH + k + 1];
        // B fragment (4x16): row K striped across lanes within a VGPR
        bf[0] = W[(size_t)k * DD + col];
        bf[1] = W[(size_t)(k + 1) * DD + col];
        accW = __builtin_amdgcn_wmma_f32_16x16x4_f32(
                   false, a, false, bf, (short)0, accW, false, false);

        a2[0] = lds_pre[m * LDS_PITCH + k];
        a2[1] = lds_pre[m * LDS_PITCH + k + 1];
        b2[0] = Wmsg[(size_t)k * DD + col];
        b2[1] = Wmsg[(size_t)(k + 1) * DD + col];
        accM = __builtin_amdgcn_wmma_f32_16x16x4_f32(
                   false, a2, false, b2, (short)0, accM, false, false);
    }

    // Epilogue: out(mRow, col) = relu(accW + accM*norm[mRow] + b[col])
    const float bb = b[col];
    if (fullTile) {
        // Common path: straight-line, no per-row EXEC churn.
        const int rowBase = nodeBase + 8 * kh;   // C layout: VGPR v -> M=v (+8 hi lanes)
        #pragma unroll
        for (int v = 0; v < 8; ++v) {
            const int row = rowBase + v;
            const float nv  = norm[row];
            float val = fmaf(accM[v], nv, accW[v]) + bb;
            out[(size_t)row * DD + col] = val > 0.f ? val : 0.f;
        }
    } else {
        #pragma unroll
        for (int v = 0; v < 8; ++v) {
            const int row = nodeBase + v + 8 * kh;
            if (row < N) {
                const float nv  = norm[row];
                float val = fmaf(accM[v], nv, accW[v]) + bb;
                out[(size_t)row * DD + col] = val > 0.f ? val : 0.f;
            }
        }
    }
}

// ---------------------------------------------------------------------------
extern "C" void kernel_launch(void* const* d_in, const int* in_sizes, int n_in,
                              void* d_out, int out_size, void* d_ws, size_t ws_size,
                              hipStream_t stream) {
    const float* h    = (const float*)d_in[0];
    const float* r    = (const float*)d_in[1];
    const float* norm = (const float*)d_in[2];
    const int*   src  = (const int*)d_in[3];
    const int*   dst  = (const int*)d_in[4];
    const int*   rel  = (const int*)d_in[5];
    const float* Wmsg = (const float*)d_in[6];
    const float* W    = (const float*)d_in[7];
    const float* b    = (const float*)d_in[8];
    float* out = (float*)d_out;

    const int N = in_sizes[0] / DD;
    const int E = in_sizes[3];

    float* hn  = (float*)d_ws;               // N*128 floats
    float* pre = hn + (size_t)N * DD;        // N*128 floats

    // 1) hn = h*norm, pre = 0
    int total4 = N * (DD / 4);
    compgcn_prep_kernel<<<(total4 + 255) / 256, 256, 0, stream>>>(h, norm, hn, pre, total4);

    // 2) edge scatter-add (linearity: segment_sum before W_msg)
    long long ethreads = (long long)E * 32;
    compgcn_edge_kernel<<<(int)((ethreads + 255) / 256), 256, 0, stream>>>(
        hn, r, src, dst, rel, pre, E);

    // 3) fused dual GEMM + norm-scale + bias + relu via v_wmma_f32_16x16x4_f32
    compgcn_gemm_kernel<<<(N + 15) / 16, 256, 0, stream>>>(
        hn, pre, norm, W, Wmsg, b, out, N);
}